// GRN_70557722738746
// MI455X (gfx1250) — compile-verified
//
#include <hip/hip_runtime.h>
#include <math.h>

typedef float v2f __attribute__((ext_vector_type(2)));
typedef float v8f __attribute__((ext_vector_type(8)));

#define BB 16
#define CC 384
#define HW 4096        // 64*64
#define NSLICE (BB*CC) // 6144
#define PITCH 65       // 64 + 1 pad -> conflict-free column access
#define PITER 20       // power iterations on G^8
#define GSCALE (1.0f/256.0f)

// ---------------------------------------------------------------------------
// Symmetric product: dst = (src * src^T) * scale, both 64x64 in padded LDS.
// 16 output tiles (4x4 of 16x16); each of 8 wave32s computes 2 tiles with
// V_WMMA_F32_16X16X4_F32. If src is symmetric this computes src^2 * scale.
// All 256 threads must execute (EXEC all-ones for WMMA).
// ---------------------------------------------------------------------------
__device__ __forceinline__ void symm_aat(const float* __restrict__ src,
                                         float* __restrict__ dst, float scale,
                                         int wave, int half, int lm) {
  for (int t = 0; t < 2; ++t) {
    const int tile = wave * 2 + t;
    const int m0 = (tile >> 2) << 4;
    const int n0 = (tile & 3) << 4;
    v8f acc = {};
#pragma unroll
    for (int kk = 0; kk < 16; ++kk) {
      // A 16x4 (MxK): VGPR j -> K = 2*half + j (lane half selects K pair)
      const int k0 = kk * 4 + 2 * half;
      v2f a, b;
      a.x = src[(m0 + lm) * PITCH + k0 + 0];
      a.y = src[(m0 + lm) * PITCH + k0 + 1];
      // B 4x16 (KxN) of src^T: B[kk][n] = src[n][kk]
      b.x = src[(n0 + lm) * PITCH + k0 + 0];
      b.y = src[(n0 + lm) * PITCH + k0 + 1];
      acc = __builtin_amdgcn_wmma_f32_16x16x4_f32(
          /*neg_a=*/false, a, /*neg_b=*/false, b,
          /*c_mod=*/(short)0, acc, /*reuse_a=*/false, /*reuse_b=*/false);
    }
    // C/D 16x16 f32 layout: VGPR r -> M = r + 8*half, N = lm
#pragma unroll
    for (int r = 0; r < 8; ++r) {
      dst[(m0 + r + 8 * half) * PITCH + n0 + lm] = acc[r] * scale;
    }
  }
}

// ---------------------------------------------------------------------------
// Kernel 1: per-slice spectral norm.
//   G' = (X X^T)/256  -> G'^2 -> G'^4 -> G'^8   (all WMMA, ping-pong LDS)
//   20 power iterations on G'^8, then sigma = 16 * lambda(G'^8)^(1/16).
// One 256-thread block (8 wave32) per (b,c) slice.
// ---------------------------------------------------------------------------
__global__ __launch_bounds__(256) void spectral_kernel(
    const float* __restrict__ x, float* __restrict__ s_out) {
  __shared__ float buf0[64 * PITCH];
  __shared__ float buf1[64 * PITCH];
  __shared__ float sv[64];
  __shared__ float sw[64];
  __shared__ float slam;

  const int tid = threadIdx.x;
  const int slice = blockIdx.x;
  const float* xs = x + (size_t)slice * HW;

  // Stage X[64][64] into padded LDS (coalesced global reads).
  for (int i = tid; i < HW; i += 256) {
    buf0[(i >> 6) * PITCH + (i & 63)] = xs[i];
  }
  __syncthreads();

  const int wave = tid >> 5;
  const int lane = tid & 31;
  const int half = lane >> 4;
  const int lm   = lane & 15;

  symm_aat(buf0, buf1, GSCALE, wave, half, lm);  // G'   = X X^T / 256
  __syncthreads();
  symm_aat(buf1, buf0, 1.0f, wave, half, lm);    // G'^2
  __syncthreads();
  symm_aat(buf0, buf1, 1.0f, wave, half, lm);    // G'^4
  __syncthreads();
  symm_aat(buf1, buf0, 1.0f, wave, half, lm);    // G'^8  (in buf0)
  __syncthreads();

  // ---- power iteration on M = G'^8 (64 lanes own one row each) ----
  if (tid < 64) sv[tid] = 1.0f;
  __syncthreads();
  for (int it = 0; it < PITER; ++it) {
    if (tid < 64) {
      float acc = 0.0f;
      const float* grow = &buf0[tid * PITCH];
#pragma unroll 8
      for (int j = 0; j < 64; ++j) acc += grow[j] * sv[j];
      sw[tid] = acc;
    }
    __syncthreads();
    if (tid < 64) {
      float n2 = 0.0f;
#pragma unroll 8
      for (int j = 0; j < 64; ++j) n2 += sw[j] * sw[j];
      const float inv = rsqrtf(n2);
      sv[tid] = sw[tid] * inv;             // v <- Mv / ||Mv||
      if (tid == 0) slam = sqrtf(n2);      // lambda_max(G'^8) estimate
    }
    __syncthreads();
  }
  if (tid == 0) {
    // sigma = sqrt(lambda_max(G)) = 16 * lambda(G'^8)^(1/16)
    float t = slam;
    t = sqrtf(t);   // ^(1/2)
    t = sqrtf(t);   // ^(1/4)
    t = sqrtf(t);   // ^(1/8)
    t = sqrtf(t);   // ^(1/16)
    s_out[slice] = 16.0f * t;
  }
}

// ---------------------------------------------------------------------------
// Kernel 2: global sum of the 6144 spectral norms.
// ---------------------------------------------------------------------------
__global__ __launch_bounds__(256) void sum_kernel(
    const float* __restrict__ s, float* __restrict__ total) {
  __shared__ float red[256];
  float acc = 0.0f;
  for (int i = threadIdx.x; i < NSLICE; i += 256) acc += s[i];
  red[threadIdx.x] = acc;
  __syncthreads();
  for (int off = 128; off > 0; off >>= 1) {
    if (threadIdx.x < off) red[threadIdx.x] += red[threadIdx.x + off];
    __syncthreads();
  }
  if (threadIdx.x == 0) total[0] = red[0];
}

// ---------------------------------------------------------------------------
// Kernel 3: out = x*(1 + gamma*s/total) + beta   (fused affine + residual)
// One block per slice; x should still be L2-resident (100MB < 192MB L2).
// ---------------------------------------------------------------------------
__global__ __launch_bounds__(256) void apply_kernel(
    const float* __restrict__ x, const float* __restrict__ gamma,
    const float* __restrict__ beta, const float* __restrict__ s,
    const float* __restrict__ total, float* __restrict__ out) {
  const int slice = blockIdx.x;
  const float sc  = s[slice] / total[0];
  const float fac = 1.0f + gamma[slice] * sc;
  const float bt  = beta[slice];
  const float4* xv = (const float4*)(x + (size_t)slice * HW);
  float4* ov       = (float4*)(out + (size_t)slice * HW);
#pragma unroll 4
  for (int i = threadIdx.x; i < HW / 4; i += 256) {
    float4 v = xv[i];
    float4 r;
    r.x = fmaf(v.x, fac, bt);
    r.y = fmaf(v.y, fac, bt);
    r.z = fmaf(v.z, fac, bt);
    r.w = fmaf(v.w, fac, bt);
    ov[i] = r;
  }
}

// ---------------------------------------------------------------------------
extern "C" void kernel_launch(void* const* d_in, const int* in_sizes, int n_in,
                              void* d_out, int out_size, void* d_ws, size_t ws_size,
                              hipStream_t stream) {
  const float* x     = (const float*)d_in[0];
  const float* gamma = (const float*)d_in[1];
  const float* beta  = (const float*)d_in[2];
  float* out = (float*)d_out;

  float* s     = (float*)d_ws;      // NSLICE floats
  float* total = s + NSLICE;        // 1 float

  spectral_kernel<<<NSLICE, 256, 0, stream>>>(x, s);
  sum_kernel<<<1, 256, 0, stream>>>(s, total);
  apply_kernel<<<NSLICE, 256, 0, stream>>>(x, gamma, beta, s, total, out);
}